// VQCodebook_61220463837584
// MI455X (gfx1250) — compile-verified
//
#include <hip/hip_runtime.h>
#include <hip/hip_bf16.h>
#include <stdint.h>

#define NUM_CLASSIFICATIONS 32
#define NUM_CLASSES         512
#define EMBED_DIMS          128
#define NUM_EMBEDDINGS      (NUM_CLASSIFICATIONS * NUM_CLASSES)   // 16384
#define BATCH               4096

typedef float v4f __attribute__((ext_vector_type(4)));

// ---------------------------------------------------------------------------
// Stage 1 (optional, needs 8 MB of workspace):
// transpose emb (EMBED_DIMS x NUM_EMBEDDINGS) -> embT (NUM_EMBEDDINGS x EMBED_DIMS)
// so the per-(b,c) gather becomes a contiguous 512B read instead of a 64KB-stride
// column walk. Stages tiles through LDS using the gfx1250 async global->LDS path
// (ASYNCcnt), per CDNA5 ISA.
// ---------------------------------------------------------------------------
__global__ __launch_bounds__(256) void vq_transpose_kernel(
    const float* __restrict__ emb, float* __restrict__ embT) {
  __shared__ float tile[32][33];                 // +1 pad: no bank conflicts
  const int tx   = threadIdx.x;                 // 0..31
  const int ty   = threadIdx.y;                 // 0..7
  const int col0 = blockIdx.x * 32;             // over NUM_EMBEDDINGS
  const int row0 = blockIdx.y * 32;             // over EMBED_DIMS

  // Async copy global -> LDS: per-lane LDS byte offset in a VGPR, 64-bit global
  // address in a VGPR pair. Tracked by ASYNCcnt.
#pragma unroll
  for (int k = 0; k < 4; ++k) {
    const int r = ty + k * 8;
    unsigned lds_off = (unsigned)(uintptr_t)&tile[r][tx];
    unsigned long long gaddr = (unsigned long long)(uintptr_t)(
        emb + (size_t)(row0 + r) * NUM_EMBEDDINGS + (size_t)(col0 + tx));
    asm volatile("global_load_async_to_lds_b32 %0, %1, off"
                 :: "v"(lds_off), "v"(gaddr)
                 : "memory");
  }
  asm volatile("s_wait_asynccnt 0" ::: "memory");
  __syncthreads();

#pragma unroll
  for (int k = 0; k < 4; ++k) {
    const int r = ty + k * 8;
    embT[(size_t)(col0 + r) * EMBED_DIMS + (size_t)(row0 + tx)] = tile[tx][r];
  }
}

// ---------------------------------------------------------------------------
// Stage 2 (fused, bandwidth-bound): one wave32 per (batch, classification).
//  - argmax over 512 fp32 classes: 4 coalesced b128 NT loads per lane,
//    wave32 butterfly reduce with first-occurrence tie-break (jnp.argmax).
//  - gather the winning 128-float embedding row (contiguous if transposed),
//    NT store of the write-once output.
// ---------------------------------------------------------------------------
__global__ __launch_bounds__(256) void vq_argmax_gather_kernel(
    const float* __restrict__ onehot,   // (BATCH, 32, 512)
    const float* __restrict__ emb,      // (128, 16384) original layout
    const float* __restrict__ embT,     // (16384, 128) transposed, or garbage
    float* __restrict__ out,            // (BATCH, 4096)
    int use_t) {
  const int lane = threadIdx.x & 31;
  const int wid  = (int)((blockIdx.x * blockDim.x + threadIdx.x) >> 5);
  const int b    = wid >> 5;                    // / NUM_CLASSIFICATIONS
  const int c    = wid & 31;                    // % NUM_CLASSIFICATIONS

  const float* src = onehot + (size_t)wid * NUM_CLASSES;

  float best = -__builtin_inff();
  int   bidx = 0;
#pragma unroll
  for (int j = 0; j < 4; ++j) {
    const int base = j * 128 + lane * 4;        // 512B contiguous per wave-load
    v4f v = __builtin_nontemporal_load((const v4f*)(src + base));
    if (v.x > best) { best = v.x; bidx = base + 0; }
    if (v.y > best) { best = v.y; bidx = base + 1; }
    if (v.z > best) { best = v.z; bidx = base + 2; }
    if (v.w > best) { best = v.w; bidx = base + 3; }
  }
  // wave32 reduction; ties -> lowest index (matches jnp.argmax first-occurrence)
#pragma unroll
  for (int off = 16; off > 0; off >>= 1) {
    float ov = __shfl_xor(best, off, 32);
    int   oi = __shfl_xor(bidx, off, 32);
    if (ov > best || (ov == best && oi < bidx)) { best = ov; bidx = oi; }
  }

  const int e = c * NUM_CLASSES + bidx;         // flat embedding index
  float* dst = out + (size_t)b * (NUM_CLASSIFICATIONS * EMBED_DIMS)
                   + (size_t)c * EMBED_DIMS;

  if (use_t) {
    // contiguous 512B wave read from L2-resident transposed table
    v4f v = *(const v4f*)(embT + (size_t)e * EMBED_DIMS + (size_t)(lane * 4));
    __builtin_nontemporal_store(v, (v4f*)(dst + lane * 4));
  } else {
    // fallback: strided column gather from original layout
#pragma unroll
    for (int d0 = 0; d0 < EMBED_DIMS; d0 += 32) {
      dst[d0 + lane] = emb[(size_t)(d0 + lane) * NUM_EMBEDDINGS + (size_t)e];
    }
  }
}

extern "C" void kernel_launch(void* const* d_in, const int* in_sizes, int n_in,
                              void* d_out, int out_size, void* d_ws, size_t ws_size,
                              hipStream_t stream) {
  (void)in_sizes; (void)n_in; (void)out_size;
  const float* onehot = (const float*)d_in[0];  // (4096, 32, 512) fp32
  const float* emb    = (const float*)d_in[1];  // (128, 16384)   fp32
  float*       out    = (float*)d_out;          // (4096, 4096)   fp32

  const size_t embT_bytes = (size_t)NUM_EMBEDDINGS * EMBED_DIMS * sizeof(float);
  float* embT = nullptr;
  int use_t = 0;
  if (ws_size >= embT_bytes) {
    embT = (float*)d_ws;
    use_t = 1;
    dim3 grid(NUM_EMBEDDINGS / 32, EMBED_DIMS / 32);  // (512, 4)
    dim3 block(32, 8);
    vq_transpose_kernel<<<grid, block, 0, stream>>>(emb, embT);
  }

  const int total_waves = BATCH * NUM_CLASSIFICATIONS;    // 131072
  const int block_sz    = 256;                            // 8 waves
  const int grid_sz     = total_waves / (block_sz / 32);  // 16384
  vq_argmax_gather_kernel<<<grid_sz, block_sz, 0, stream>>>(
      onehot, emb, embT ? embT : emb, out, use_t);
}